// DMT_14027363188880
// MI455X (gfx1250) — compile-verified
//
#include <hip/hip_runtime.h>
#include <hip/hip_bf16.h>

typedef __attribute__((ext_vector_type(2))) float v2f;
typedef __attribute__((ext_vector_type(8))) float v8f;

constexpr int M_  = 4096;
constexpr int B_  = 8;
constexpr int N_  = 512;
constexpr int FD_ = 128;
constexpr int BD_ = 128;
constexpr int BVD_= 6;
constexpr int H_  = 64;
constexpr int H2_ = 32;

// D = A(16x4) * B(4x16) + C  via V_WMMA_F32_16X16X4_F32
__device__ __forceinline__ v8f wmma4(v2f a, v2f b, v8f c) {
  return __builtin_amdgcn_wmma_f32_16x16x4_f32(
      /*neg_a=*/false, a, /*neg_b=*/false, b,
      /*c_mod=*/(short)0, c, /*reuse_a=*/false, /*reuse_b=*/false);
}

// ---------------------------------------------------------------------------
// Kernel 1: fT[m][h] = sum_d f_pre_in[m][d] * Wf[h][d] + bf[h]   (M x H)
// ---------------------------------------------------------------------------
__global__ __launch_bounds__(32)
void fT_kernel(const float* __restrict__ fpre, const float* __restrict__ Wf,
               const float* __restrict__ bf, float* __restrict__ fT) {
  const int m0 = blockIdx.x * 16;
  const int h0 = blockIdx.y * 16;
  const int lane = threadIdx.x, r = lane & 15, s = lane >> 4;
  v8f c;
  const float bias = bf[h0 + r];            // bias per output column h
  #pragma unroll
  for (int i = 0; i < 8; ++i) c[i] = bias;
  #pragma unroll 8
  for (int k = 0; k < FD_; k += 4) {
    v2f a, b;
    const float* ap = fpre + (size_t)(m0 + r) * FD_ + k + 2 * s;
    a.x = ap[0]; a.y = ap[1];
    const float* bp = Wf + (size_t)(h0 + r) * FD_ + k + 2 * s;  // B[k][n]=Wf[n][k]
    b.x = bp[0]; b.y = bp[1];
    c = wmma4(a, b, c);
  }
  #pragma unroll
  for (int i = 0; i < 8; ++i)
    fT[(size_t)(m0 + i + 8 * s) * H_ + h0 + r] = c[i];
}

// ---------------------------------------------------------------------------
// Kernel 2: b_feat[b][h][n] = sum_d Wb[h][d] * b_pre_in[b][d][n] + bb[h]
// ---------------------------------------------------------------------------
__global__ __launch_bounds__(32)
void bfeat_kernel(const float* __restrict__ bpre, const float* __restrict__ Wb,
                  const float* __restrict__ bb, float* __restrict__ b_feat) {
  const int n0 = blockIdx.x * 16;
  const int h0 = blockIdx.y * 16;
  const int b  = blockIdx.z;
  const int lane = threadIdx.x, r = lane & 15, s = lane >> 4;
  v8f c;
  #pragma unroll
  for (int i = 0; i < 8; ++i) c[i] = bb[h0 + i + 8 * s];  // bias per output row h
  const float* bp = bpre + (size_t)b * BD_ * N_;
  #pragma unroll 8
  for (int k = 0; k < BD_; k += 4) {
    v2f a, bfr;
    const float* ap = Wb + (size_t)(h0 + r) * BD_ + k + 2 * s;  // A rows = h
    a.x = ap[0]; a.y = ap[1];
    bfr.x = bp[(size_t)(k + 2 * s)     * N_ + n0 + r];
    bfr.y = bp[(size_t)(k + 2 * s + 1) * N_ + n0 + r];
    c = wmma4(a, bfr, c);
  }
  #pragma unroll
  for (int i = 0; i < 8; ++i)
    b_feat[((size_t)b * H_ + h0 + i + 8 * s) * N_ + n0 + r] = c[i];
}

// ---------------------------------------------------------------------------
// Kernel 3: bv_feat[b][h][n] = sum_c Wbv[h][c] * bv_in[b][c][n] + bbv[h]  (K=6)
// ---------------------------------------------------------------------------
__global__ __launch_bounds__(256)
void bvfeat_kernel(const float* __restrict__ bv_in, const float* __restrict__ Wbv,
                   const float* __restrict__ bbv, float* __restrict__ bv_feat) {
  const int idx = blockIdx.x * 256 + threadIdx.x;   // over B*H*N
  const int n   = idx & (N_ - 1);
  const int h   = (idx / N_) & (H_ - 1);
  const int b   = idx / (N_ * H_);
  float acc = bbv[h];
  #pragma unroll
  for (int cc = 0; cc < BVD_; ++cc)
    acc += Wbv[h * BVD_ + cc] * bv_in[((size_t)b * BVD_ + cc) * N_ + n];
  bv_feat[idx] = acc;
}

// ---------------------------------------------------------------------------
// Kernel 4: fused attention tile. 4 waves (128 threads) handle 16 rows of m.
//   Wave w: logits col-tiles nt = w, w+4, ... ; AV out-tile ht = w; proj tile = w.
//   logits = fT_tile[:, hb:hb+32] @ b_feat[b][hb:hb+32, :]   (row-selected by batch)
//   w = softmax(logits / 8); out_phase = w_masked @ bv_feat[b]^T (accum over b)
//   cat = [out_euc | out_geo]; out = cat @ Wo^T + bo
// ---------------------------------------------------------------------------
__global__ __launch_bounds__(128)
void attn_kernel(const int* __restrict__ fbatch, const float* __restrict__ fT,
                 const float* __restrict__ b_feat, const float* __restrict__ bv_feat,
                 const float* __restrict__ Wo, const float* __restrict__ bo,
                 float* __restrict__ out) {
  __shared__ float w_buf[16][N_];        // 32 KB: logits, then softmax weights
  __shared__ float cat_buf[16][2 * H_];  // 8 KB : [out_euc | out_geo]
  __shared__ int   rowb[16];

  const int m0   = blockIdx.x * 16;
  const int tid  = threadIdx.x;
  const int wave = tid >> 5;
  const int lane = tid & 31, r = lane & 15, s = lane >> 4;

  if (tid < 16) rowb[tid] = fbatch[m0 + tid];
  __syncthreads();
  const int b_lo = __builtin_amdgcn_readfirstlane(rowb[0]);
  const int b_hi = __builtin_amdgcn_readfirstlane(rowb[15]);  // sorted => span
  const int myb  = rowb[r];      // batch of A-matrix row r for this lane
  int rowvals[8];                // batch id of each C-matrix row this lane owns
  #pragma unroll
  for (int i = 0; i < 8; ++i) rowvals[i] = rowb[i + 8 * s];

  for (int phase = 0; phase < 2; ++phase) {
    const int hb = phase * H2_;
    // A-fragments for the whole 16x32 fT slice: 8 k-steps resident in regs
    v2f af[8];
    {
      const float* frow = fT + (size_t)(m0 + r) * H_ + hb + 2 * s;
      #pragma unroll
      for (int kk = 0; kk < 8; ++kk) { af[kk].x = frow[4 * kk]; af[kk].y = frow[4 * kk + 1]; }
    }
    // ---- logits (WMMA); waves split the 32 column tiles ----
    for (int nt = wave; nt < N_ / 16; nt += 4) {
      for (int b = b_lo; b <= b_hi; ++b) {
        const float* bf_b = b_feat + ((size_t)b * H_ + hb) * N_;
        v8f c;
        #pragma unroll
        for (int i = 0; i < 8; ++i) c[i] = 0.0f;
        #pragma unroll
        for (int kk = 0; kk < 8; ++kk) {
          v2f bfr;
          bfr.x = bf_b[(size_t)(4 * kk + 2 * s)     * N_ + nt * 16 + r];
          bfr.y = bf_b[(size_t)(4 * kk + 2 * s + 1) * N_ + nt * 16 + r];
          c = wmma4(af[kk], bfr, c);
        }
        #pragma unroll
        for (int i = 0; i < 8; ++i)
          if (rowvals[i] == b) w_buf[i + 8 * s][nt * 16 + r] = c[i];
      }
    }
    __syncthreads();
    // ---- softmax over n (scale 1/sqrt(H)=0.125); 8 lanes x 64 cols per row ----
    {
      const int row = 4 * wave + (lane & 3);   // rows in lane bits [1:0]
      const int base = (lane >> 2) * 64;       // segment in lane bits [4:2]
      float mx = -3.4e38f;
      for (int n = 0; n < 64; ++n) mx = fmaxf(mx, w_buf[row][base + n]);
      mx = fmaxf(mx, __shfl_xor(mx, 4, 32));
      mx = fmaxf(mx, __shfl_xor(mx, 8, 32));
      mx = fmaxf(mx, __shfl_xor(mx, 16, 32));
      float sum = 0.0f;
      for (int n = 0; n < 64; ++n) {
        const float e = __expf((w_buf[row][base + n] - mx) * 0.125f);
        w_buf[row][base + n] = e;
        sum += e;
      }
      sum += __shfl_xor(sum, 4, 32);
      sum += __shfl_xor(sum, 8, 32);
      sum += __shfl_xor(sum, 16, 32);
      const float inv = 1.0f / sum;
      for (int n = 0; n < 64; ++n) w_buf[row][base + n] *= inv;
    }
    __syncthreads();
    // ---- out_phase = w_masked @ bv_feat[b]^T; wave w owns out-tile ht = w ----
    {
      const int ht = wave;
      v8f c;
      #pragma unroll
      for (int i = 0; i < 8; ++i) c[i] = 0.0f;
      for (int b = b_lo; b <= b_hi; ++b) {
        const float* bv_b = bv_feat + (size_t)b * H_ * N_ + (size_t)(ht * 16 + r) * N_;
        const bool rowon = (myb == b);
        #pragma unroll 4
        for (int k = 0; k < N_; k += 4) {
          v2f a, bfr;
          a.x = rowon ? w_buf[r][k + 2 * s]     : 0.0f;
          a.y = rowon ? w_buf[r][k + 2 * s + 1] : 0.0f;
          bfr.x = bv_b[k + 2 * s];              // B[k][h] = bv_feat[b][h][k]
          bfr.y = bv_b[k + 2 * s + 1];
          c = wmma4(a, bfr, c);
        }
      }
      #pragma unroll
      for (int i = 0; i < 8; ++i)
        cat_buf[i + 8 * s][phase * H_ + ht * 16 + r] = c[i];
    }
    __syncthreads();   // also guards w_buf reuse by next phase
  }
  // ---- out tile = cat(16x128) @ Wo^T(128x64) + bo; wave w owns tile ht = w ----
  {
    const int ht = wave;
    v8f c;
    const float bias = bo[ht * 16 + r];
    #pragma unroll
    for (int i = 0; i < 8; ++i) c[i] = bias;
    #pragma unroll 4
    for (int k = 0; k < 2 * H_; k += 4) {
      v2f a, bfr;
      a.x = cat_buf[r][k + 2 * s];
      a.y = cat_buf[r][k + 2 * s + 1];
      const float* bp = Wo + (size_t)(ht * 16 + r) * (2 * H_) + k + 2 * s;
      bfr.x = bp[0]; bfr.y = bp[1];             // B[k][h] = Wo[h][k]
      c = wmma4(a, bfr, c);
    }
    #pragma unroll
    for (int i = 0; i < 8; ++i)
      out[(size_t)(m0 + i + 8 * s) * H_ + ht * 16 + r] = c[i];
  }
}

// ---------------------------------------------------------------------------
extern "C" void kernel_launch(void* const* d_in, const int* in_sizes, int n_in,
                              void* d_out, int out_size, void* d_ws, size_t ws_size,
                              hipStream_t stream) {
  const float* f_pre_in    = (const float*)d_in[0];
  const int*   f_pre_batch = (const int*)  d_in[1];
  const float* b_pre_in    = (const float*)d_in[2];
  const float* bv_in       = (const float*)d_in[3];
  const float* Wf          = (const float*)d_in[4];
  const float* bf          = (const float*)d_in[5];
  const float* Wb          = (const float*)d_in[6];
  const float* bb          = (const float*)d_in[7];
  const float* Wbv         = (const float*)d_in[8];
  const float* bbv         = (const float*)d_in[9];
  const float* Wo          = (const float*)d_in[10];
  const float* bo          = (const float*)d_in[11];
  float* out = (float*)d_out;

  float* ws      = (float*)d_ws;
  float* fT      = ws;                               // M*H      = 256K floats
  float* b_feat  = fT + (size_t)M_ * H_;             // B*H*N    = 256K floats
  float* bv_feat = b_feat + (size_t)B_ * H_ * N_;    // B*H*N    = 256K floats

  fT_kernel    <<<dim3(M_ / 16, H_ / 16),      32, 0, stream>>>(f_pre_in, Wf, bf, fT);
  bfeat_kernel <<<dim3(N_ / 16, H_ / 16, B_),  32, 0, stream>>>(b_pre_in, Wb, bb, b_feat);
  bvfeat_kernel<<<dim3((B_ * H_ * N_) / 256), 256, 0, stream>>>(bv_in, Wbv, bbv, bv_feat);
  attn_kernel  <<<dim3(M_ / 16),              128, 0, stream>>>(f_pre_batch, fT, b_feat,
                                                                bv_feat, Wo, bo, out);
}